// GraphModel_31275951850041
// MI455X (gfx1250) — compile-verified
//
#include <hip/hip_runtime.h>

// MI455X / gfx1250, wave32. H = 64 hidden features throughout.
#define HF 64

typedef float v2f __attribute__((ext_vector_type(2)));
typedef float v8f __attribute__((ext_vector_type(8)));

// ---------------------------------------------------------------------------
// Scatter-add: agg[dst[e], :] += h[src[e], :]  (edge-parallel, float4 gather,
// hardware f32 atomics -> resolves in L2; whole working set is L2-resident)
// ---------------------------------------------------------------------------
__global__ void scatter_add_kernel(const float* __restrict__ h,
                                   const int* __restrict__ src,
                                   const int* __restrict__ dst,
                                   float* __restrict__ agg,
                                   int nEdges) {
  int tid  = blockIdx.x * blockDim.x + threadIdx.x;
  int edge = tid >> 4;             // 16 threads per edge, 4 floats each
  if (edge >= nEdges) return;
  int f = (tid & 15) << 2;
  int s = src[edge];
  int d = dst[edge];
  const float4 v = *(const float4*)(h + (size_t)s * HF + f);
  float* p = agg + (size_t)d * HF + f;
  unsafeAtomicAdd(p + 0, v.x);
  unsafeAtomicAdd(p + 1, v.y);
  unsafeAtomicAdd(p + 2, v.z);
  unsafeAtomicAdd(p + 3, v.w);
}

// ---------------------------------------------------------------------------
// Fused GraphConv GEMM + ReLU via fp32 WMMA:
//   out = relu( agg @ Wrel^T + h @ Wroot^T )
// One wave per 16-row tile; K=64 as 16 chunks of V_WMMA_F32_16X16X4_F32.
// fp32 A(16x4): lane m = lane&15; lanes 0-15 hold K={0,1}, 16-31 K={2,3}
// fp32 B(4x16): mirrored -> both operands are a float2 at row*64 + k.
// ---------------------------------------------------------------------------
__global__ void gemm_relu_kernel(const float* __restrict__ agg,
                                 const float* __restrict__ h,
                                 const float* __restrict__ Wrel,
                                 const float* __restrict__ Wroot,
                                 float* __restrict__ out,
                                 int nTiles) {
  int wave = (blockIdx.x * blockDim.x + threadIdx.x) >> 5; // global wave id
  int lane = threadIdx.x & 31;
  if (wave >= nTiles) return;      // wave-uniform: EXEC stays all-ones inside
  const int m    = lane & 15;
  const int half = lane >> 4;      // 0: K lo pair, 1: K hi pair
  const float* aggRow = agg + ((size_t)wave * 16 + m) * HF;
  const float* hRow   = h   + ((size_t)wave * 16 + m) * HF;

  for (int nt = 0; nt < 4; ++nt) {           // 4 N-tiles -> full 64 outputs
    const float* wrelRow  = Wrel  + (size_t)(nt * 16 + m) * HF;
    const float* wrootRow = Wroot + (size_t)(nt * 16 + m) * HF;
    v8f acc = {};
    for (int kt = 0; kt < 16; ++kt) {
      const int k = kt * 4 + half * 2;
      v2f aA = *(const v2f*)(aggRow + k);
      v2f bA = *(const v2f*)(wrelRow + k);
      acc = __builtin_amdgcn_wmma_f32_16x16x4_f32(
          /*neg_a=*/false, aA, /*neg_b=*/false, bA,
          /*c_mod=*/(short)0, acc, /*reuse_a=*/false, /*reuse_b=*/false);
      v2f aH = *(const v2f*)(hRow + k);
      v2f bH = *(const v2f*)(wrootRow + k);
      acc = __builtin_amdgcn_wmma_f32_16x16x4_f32(
          false, aH, false, bH, (short)0, acc, false, false);
    }
    // D layout: VGPR j -> row (j + 8*half), col = lane&15 ; apply ReLU.
#pragma unroll
    for (int j = 0; j < 8; ++j) {
      float v = acc[j];
      v = v < 0.0f ? 0.0f : v;
      out[((size_t)wave * 16 + j + 8 * half) * HF + nt * 16 + m] = v;
    }
  }
}

// ---------------------------------------------------------------------------
// Global mean-pool accumulation: sums[batch[i],:] += h[i,:]; counts[g] += 1
// ---------------------------------------------------------------------------
__global__ void pool_kernel(const float* __restrict__ h,
                            const int* __restrict__ batch,
                            float* __restrict__ sums,
                            float* __restrict__ counts,
                            int nNodes) {
  int tid  = blockIdx.x * blockDim.x + threadIdx.x;
  int node = tid >> 6;
  if (node >= nNodes) return;
  int f = tid & 63;
  int g = batch[node];
  unsafeAtomicAdd(&sums[(size_t)g * HF + f], h[(size_t)node * HF + f]);
  if (f == 0) unsafeAtomicAdd(&counts[g], 1.0f);
}

// ---------------------------------------------------------------------------
// Output projection via WMMA: out[128,32] = (sums/counts) @ Wout^T
// 8 M-tiles x 2 N-tiles = 16 waves in a single 512-thread block.
// Mean division folded into the A-operand load (per-lane row is fixed).
// ---------------------------------------------------------------------------
__global__ void final_gemm_kernel(const float* __restrict__ sums,
                                  const float* __restrict__ counts,
                                  const float* __restrict__ Wout,
                                  float* __restrict__ out) {
  const int wave = threadIdx.x >> 5;   // 0..15
  const int lane = threadIdx.x & 31;
  const int mt = wave >> 1;            // 0..7
  const int nt = wave & 1;             // 0..1
  const int m    = lane & 15;
  const int half = lane >> 4;
  const int row  = mt * 16 + m;
  const float cnt = counts[row];
  const float inv = 1.0f / (cnt < 1.0f ? 1.0f : cnt);
  const float* sRow = sums + (size_t)row * HF;
  const float* wRow = Wout + (size_t)(nt * 16 + m) * HF;

  v8f acc = {};
  for (int kt = 0; kt < 16; ++kt) {
    const int k = kt * 4 + half * 2;
    v2f a;
    a.x = sRow[k] * inv;
    a.y = sRow[k + 1] * inv;
    v2f b = *(const v2f*)(wRow + k);
    acc = __builtin_amdgcn_wmma_f32_16x16x4_f32(
        false, a, false, b, (short)0, acc, false, false);
  }
#pragma unroll
  for (int j = 0; j < 8; ++j) {
    out[(size_t)(mt * 16 + j + 8 * half) * 32 + nt * 16 + m] = acc[j];
  }
}

// ---------------------------------------------------------------------------
// Host orchestration (graph-capture safe: kernels + hipMemsetAsync only)
// ---------------------------------------------------------------------------
extern "C" void kernel_launch(void* const* d_in, const int* in_sizes, int n_in,
                              void* d_out, int out_size, void* d_ws, size_t ws_size,
                              hipStream_t stream) {
  const float* x      = (const float*)d_in[0];            // [N,64]
  const int*   eidx   = (const int*)d_in[1];              // [2,E] (int32 per harness)
  const int*   batch  = (const int*)d_in[2];              // [N]
  const float* W_rel  = (const float*)d_in[3];            // [3,64,64]
  const float* W_root = (const float*)d_in[4];            // [3,64,64]
  const float* W_out  = (const float*)d_in[5];            // [32,64]
  float* out = (float*)d_out;                             // [128,32]

  const int nNodes = in_sizes[0] / HF;                    // 100000
  const int nEdges = in_sizes[1] / 2;                     // 1600000
  const int nGraphs = out_size / 32;                      // 128
  const int*   src = eidx;
  const int*   dst = eidx + nEdges;

  // Workspace carve-up
  char* ws = (char*)d_ws;
  const size_t hbytes = (size_t)nNodes * HF * sizeof(float);  // 25.6 MB
  float* agg    = (float*)(ws);
  float* hA     = (float*)(ws + hbytes);
  float* hB     = (float*)(ws + 2 * hbytes);
  float* sums   = (float*)(ws + 3 * hbytes);
  float* counts = (float*)(ws + 3 * hbytes + (size_t)nGraphs * HF * sizeof(float));
  (void)ws_size; (void)n_in;

  const int nTiles = nNodes / 16;                         // 6250 exact
  dim3 blk(256);
  dim3 gridScatter((unsigned)(((size_t)nEdges * 16 + 255) / 256));
  dim3 gridGemm((unsigned)((nTiles + 7) / 8));            // 8 waves / block
  dim3 gridPool((unsigned)(((size_t)nNodes * HF + 255) / 256));

  const float* hin = x;
  float* hout[3] = {hA, hB, hA};
  for (int l = 0; l < 3; ++l) {
    hipMemsetAsync(agg, 0, hbytes, stream);
    scatter_add_kernel<<<gridScatter, blk, 0, stream>>>(hin, src, dst, agg, nEdges);
    gemm_relu_kernel<<<gridGemm, blk, 0, stream>>>(
        agg, hin, W_rel + (size_t)l * HF * HF, W_root + (size_t)l * HF * HF,
        hout[l], nTiles);
    hin = hout[l];
  }

  hipMemsetAsync(sums, 0, (size_t)nGraphs * (HF + 1) * sizeof(float), stream);
  pool_kernel<<<gridPool, blk, 0, stream>>>(hin, batch, sums, counts, nNodes);

  final_gemm_kernel<<<1, dim3(512), 0, stream>>>(sums, counts, W_out, out);
}